// CanonicalCov1D_36189394436372
// MI455X (gfx1250) — compile-verified
//
#include <hip/hip_runtime.h>

// CanonicalCov1D fused WMMA kernel for gfx1250 (MI455X) — round 3.
//
// cov[b,n,c] = (1/32)*sum_l p1*p2 - mean_l(p1)*mean_l(p2) + bias[c]
//   p1[l,c] = sum_w X[b, n*8 + w]       * W1[w,l,c]
//   p2[l,c] = sum_w X[b, n*8 + 128 + w] * W2[w,l,c]
//
// 256 threads = 8 waves per WG: wave -> (c block (wave&3)*16, window subtile
// wave>>2); the two 16-window subtiles share the LDS-staged weight slices,
// halving L2 weight traffic vs one-tile WGs.  A operands (window samples)
// are register-resident per wave.  B operands are double-buffered in LDS at
// half-l-slice granularity (32 KB per buffer, 64 KB total -> 2 WGs + 4
// waves/SIMD co-resident): while WMMA chains consume buffer[h] via ds_load,
// the next half-slice is prefetched with GLOBAL_LOAD_ASYNC_TO_LDS_B128
// (ASYNCcnt), so global latency stays hidden behind compute.

typedef float v2f __attribute__((ext_vector_type(2)));
typedef float v8f __attribute__((ext_vector_type(8)));

#define N_WIN   993                 // (8064 - 128)/8 + 1
#define T_LEN   8192
#define C_DIM   64
#define LAT     32
#define ROWSTR  (LAT * C_DIM)       // 2048 floats between consecutive w rows
#define HSLICE  (64 * C_DIM)        // 4096 floats = 16 KB: half of W[:,l,:]
#define BUFSZ   (2 * HSLICE)        // W1 half + W2 half per buffer (32 KB)
#define SMEM_BYTES (2 * BUFSZ * 4)  // double buffer: 64 KB (LDS cap 320 KB)

__device__ __forceinline__ void async_ld_b128(unsigned lds_byte_off, const float* g)
{
    // GLOBAL_LOAD_ASYNC_TO_LDS_B128: vdst = per-lane LDS byte offset,
    // vaddr = 64-bit global address, no saddr.  Tracked by ASYNCcnt.
    asm volatile("global_load_async_to_lds_b128 %0, %1, off"
                 :: "v"(lds_byte_off), "v"(g) : "memory");
}

__device__ __forceinline__ void wait_async0()
{
    asm volatile("s_wait_asynccnt 0" ::: "memory");
}

__global__ __launch_bounds__(256) void
cov1d_wmma_f32_kernel(const float* __restrict__ X,
                      const float* __restrict__ W1,
                      const float* __restrict__ W2,
                      const float* __restrict__ bias,
                      float* __restrict__ out)
{
    extern __shared__ float smem[];
    // Low 32 bits of a flat pointer into LDS == wave-relative LDS byte offset.
    const unsigned sbase = (unsigned)(unsigned long long)&smem[0];

    const int tid   = threadIdx.x;
    const int lane  = tid & 31;
    const int wave  = tid >> 5;              // 8 waves
    const int c0    = (wave & 3) * 16;       // c block [c0, c0+16)
    const int sub   = wave >> 2;             // window subtile 0/1
    const int n0    = blockIdx.x * 32 + sub * 16;
    const int b     = blockIdx.y;            // batch

    const int mlane = lane & 15;             // M row (window within tile)
    const int hi    = lane >> 4;             // 0: K pair {0,1}; 1: K pair {2,3}

    // Clamp window index (loads stay in-bounds; duplicates masked at store).
    const int n_eff = min(n0 + mlane, N_WIN - 1);

    // ---- Preload A into registers ----------------------------------------
    // A-matrix 16x4 f32 layout: lanes 0-15 hold (M=lane, K=4k+{0,1}) in
    // v0/v1, lanes 16-31 hold (M=lane-16, K=4k+{2,3}).  The win2 matrix is
    // the same stream shifted by 128 samples, i.e. a[k+32].
    const float* Xrow = X + b * T_LEN + n_eff * 8 + 2 * hi;
    v2f a[64];
#pragma unroll
    for (int k = 0; k < 64; ++k)
        a[k] = *(const v2f*)(Xrow + 4 * k);   // aligned 8B pair

    // ---- Cooperative async staging of one half-slice ---------------------
    // Buffer layout: [w_local(64)][c(64)] row-major, W1 at +0, W2 at +HSLICE.
    // Buffer index == k-half index h.  256 threads x 8 B128 = 32 KB / chunk.
    const int w0 = tid >> 4;                 // 0..15
    const int c4 = tid & 15;                 // float4 column
    const float*   g1base = W1 + w0 * ROWSTR + c4 * 4;
    const float*   g2base = W2 + w0 * ROWSTR + c4 * 4;
    const unsigned lbase  = sbase + (unsigned)(w0 * C_DIM + c4 * 4) * 4u;

    auto stage_half = [&](int l, int h) {
        const float*   g1 = g1base + h * 64 * ROWSTR + l * C_DIM;
        const float*   g2 = g2base + h * 64 * ROWSTR + l * C_DIM;
        const unsigned lo = lbase + (unsigned)(h * BUFSZ) * 4u;
#pragma unroll
        for (int r = 0; r < 4; ++r) {        // 16 w-rows per round
            async_ld_b128(lo + r * 1024u * 4u,               g1 + r * 16 * ROWSTR);
            async_ld_b128(lo + r * 1024u * 4u + HSLICE * 4u, g2 + r * 16 * ROWSTR);
        }
    };

    stage_half(0, 0);
    wait_async0();
    __syncthreads();

    v8f s12 = {}; v8f s1 = {}; v8f s2 = {};

    // Per-lane base into a staged buffer.  B-matrix 4x16 layout (mirrors A's
    // K split): v0 lanes0-15 = row 4k+0, lanes16-31 = row 4k+2; v1 = +1 row.
    const int sboff = hi * 128 + c0 + mlane;

    for (int l = 0; l < LAT; ++l) {
        v8f d1 = {}; v8f d2 = {};

        // ---- half 0: compute from buf0, prefetch (l, half 1) into buf1 ---
        stage_half(l, 1);
        {
            const float* s1p = smem + sboff;            // buf0
            const float* s2p = s1p + HSLICE;
#pragma unroll
            for (int kl = 0; kl < 16; ++kl) {
                v2f b1, b2;
                b1.x = s1p[kl * 256];
                b1.y = s1p[kl * 256 + 64];
                b2.x = s2p[kl * 256];
                b2.y = s2p[kl * 256 + 64];
                d1 = __builtin_amdgcn_wmma_f32_16x16x4_f32(
                         false, a[kl],      false, b1, (short)0, d1, false, false);
                d2 = __builtin_amdgcn_wmma_f32_16x16x4_f32(
                         false, a[kl + 32], false, b2, (short)0, d2, false, false);
            }
        }
        wait_async0();
        __syncthreads();

        // ---- half 1: compute from buf1, prefetch (l+1, half 0) into buf0 -
        if (l + 1 < LAT)
            stage_half(l + 1, 0);
        {
            const float* s1p = smem + BUFSZ + sboff;    // buf1
            const float* s2p = s1p + HSLICE;
#pragma unroll
            for (int kl = 0; kl < 16; ++kl) {
                v2f b1, b2;
                b1.x = s1p[kl * 256];
                b1.y = s1p[kl * 256 + 64];
                b2.x = s2p[kl * 256];
                b2.y = s2p[kl * 256 + 64];
                d1 = __builtin_amdgcn_wmma_f32_16x16x4_f32(
                         false, a[kl + 16], false, b1, (short)0, d1, false, false);
                d2 = __builtin_amdgcn_wmma_f32_16x16x4_f32(
                         false, a[kl + 48], false, b2, (short)0, d2, false, false);
            }
        }
        s1  += d1;
        s2  += d2;
        s12 += d1 * d2;

        wait_async0();
        __syncthreads();
    }

    // ---- Epilogue: cov = s12/32 - (s1/32)(s2/32) + bias ------------------
    // C/D 16x16 f32 layout: VGPR r, lanes0-15 -> row M=r, lanes16-31 -> M=r+8.
    const float inv = 1.0f / (float)LAT;
    const float bv  = bias[c0 + mlane];
#pragma unroll
    for (int r = 0; r < 8; ++r) {
        const int n = n0 + r + 8 * hi;
        if (n < N_WIN) {
            const float m1 = s1[r] * inv;
            const float m2 = s2[r] * inv;
            out[(b * N_WIN + n) * C_DIM + c0 + mlane] = s12[r] * inv - m1 * m2 + bv;
        }
    }
}

extern "C" void kernel_launch(void* const* d_in, const int* in_sizes, int n_in,
                              void* d_out, int out_size, void* d_ws, size_t ws_size,
                              hipStream_t stream)
{
    const float* X    = (const float*)d_in[0];
    const float* W1   = (const float*)d_in[1];
    const float* W2   = (const float*)d_in[2];
    const float* bias = (const float*)d_in[3];
    float* out        = (float*)d_out;

    const int n_tiles = (N_WIN + 31) / 32;   // 32 (two 16-window subtiles/WG)
    dim3 grid(n_tiles, 32);                  // (window tile pairs, batches)
    cov1d_wmma_f32_kernel<<<grid, 256, SMEM_BYTES, stream>>>(X, W1, W2, bias, out);
}